// TopoContrastLoss_10874857193967
// MI455X (gfx1250) — compile-verified
//
#include <hip/hip_runtime.h>
#include <hip/hip_bf16.h>

// ---------------------------------------------------------------------------
// TopoContrastLoss for MI455X (gfx1250, wave32, WMMA)
//
// Per (graph, view) workgroup: stage H (256x128 f32) in LDS, compute the
// Gram matrix with V_WMMA_F32_16X16X4_F32 over upper-triangle 16x16 tiles,
// derive pairwise distances, build the 16-bin Gaussian soft histogram.
// Pass 1 gets mean(D); pass 2 recomputes tiles (cheaper than 256KB of D
// storage) and accumulates the histogram. Second kernel: signature MSE +
// NT-Xent -> scalar loss.
//
// v_sqrt_f32 (hardware sqrt, ~1 ulp) replaces the IEEE sqrt expansion:
// distances feed a sigma=0.1875 Gaussian histogram, so 1 ulp is plenty,
// and it removes the dominant VALU fixup chain from the tile epilogue.
// ---------------------------------------------------------------------------

typedef __attribute__((ext_vector_type(2))) float v2f;
typedef __attribute__((ext_vector_type(8))) float v8f;

#define NUM_GRAPHS 128
#define NPG        256      // nodes per graph
#define DF         128      // feature dim
#define KB         16       // histogram bins
#define LP         132      // LDS row pitch (floats): 132 % 64 = 4 -> conflict-free-ish
#define N_TILES    136      // upper triangle of 16x16 tile grid
#define LAMBDA_F   0.1f
#define EPS_F      1e-8f

__device__ __forceinline__ float fast_sqrtf(float x) {
  return __builtin_amdgcn_sqrtf(x);       // v_sqrt_f32
}

// Gram tile via fp32 WMMA: A = rows [tm*16, tm*16+16) of H, B = rows of H
// used as columns (Gram => B fragment has identical per-lane addressing).
// ISA layout (32-bit A 16x4): lanes 0-15 -> M=lane, K={0,1} in v0/v1;
// lanes 16-31 -> M=lane-16, K={2,3}. B (4x16) mirrors with N=lane.
__device__ __forceinline__ v8f gram_tile(const float* __restrict__ Hs,
                                         int tm, int tn, int lo, int hi) {
  const float* arow = Hs + (tm * 16 + lo) * LP + 2 * hi;
  const float* brow = Hs + (tn * 16 + lo) * LP + 2 * hi;
  v8f c = {};
#pragma unroll 8
  for (int kk = 0; kk < DF; kk += 4) {
    v2f a = *(const v2f*)(arow + kk);
    v2f b = *(const v2f*)(brow + kk);
    c = __builtin_amdgcn_wmma_f32_16x16x4_f32(
        /*neg_a=*/false, a, /*neg_b=*/false, b,
        /*c_mod=*/(short)0, c, /*reuse_a=*/false, /*reuse_b=*/false);
  }
  return c;
}

__device__ __forceinline__ void tile_coords(int idx, int& tm, int& tn) {
  int L = idx, m = 0, rl = 16;
  while (L >= rl) { L -= rl; --rl; ++m; }
  tm = m; tn = m + L;
}

__global__ void uts_sig_kernel(const float* __restrict__ H1,
                               const float* __restrict__ H2,
                               float* __restrict__ sig) {
  extern __shared__ char smem_raw[];
  float* Hs  = (float*)smem_raw;          // NPG * LP
  float* sq  = Hs + NPG * LP;             // NPG
  float* red = sq + NPG;                  // [0]=sumD, [1..16]=hist

  const int b    = blockIdx.x;
  const int view = b >> 7;
  const int g    = b & (NUM_GRAPHS - 1);
  const float* Hsrc = (view ? H2 : H1) + (size_t)g * NPG * DF;

  const int tid  = threadIdx.x;
  const int wave = tid >> 5;
  const int lane = tid & 31;
  const int hi   = lane >> 4;
  const int lo   = lane & 15;

  // ---- stage H into LDS (float4 = global_load_b128 / ds_store_b128) ----
#pragma unroll
  for (int it = 0; it < (NPG * DF) / (4 * 256); ++it) {
    int idx4 = tid + it * 256;              // float4 index
    int row  = idx4 >> 5;                   // 32 float4 per row
    int c4   = (idx4 & 31) << 2;
    float4 v = ((const float4*)Hsrc)[idx4];
    *(float4*)(Hs + row * LP + c4) = v;     // 528B pitch keeps 16B alignment
  }
  if (tid <= KB) red[tid] = 0.0f;
  __syncthreads();

  // ---- per-node squared norms ----
  {
    const float* r = Hs + tid * LP;
    float s = 0.0f;
#pragma unroll 8
    for (int k = 0; k < DF; ++k) s += r[k] * r[k];
    sq[tid] = s;
  }
  __syncthreads();

  // ---- pass 1: sum of distances (upper-triangle tiles, x2 off-diag) ----
  float sumD = 0.0f;
  for (int idx = wave; idx < N_TILES; idx += 8) {
    int tm, tn; tile_coords(idx, tm, tn);
    v8f c = gram_tile(Hs, tm, tn, lo, hi);
    const float w   = (tm == tn) ? 1.0f : 2.0f;
    const float sqj = sq[tn * 16 + lo];
#pragma unroll
    for (int v = 0; v < 8; ++v) {
      float sqi = sq[tm * 16 + v + 8 * hi];
      float d2  = sqi + sqj - 2.0f * c[v];
      float d   = fast_sqrtf(fmaxf(d2, 0.0f) + 1e-12f);
      sumD += w * d;
    }
  }
  atomicAdd(&red[0], sumD);
  __syncthreads();

  const float meanD = red[0] * (1.0f / (float)(NPG * NPG));
  const float rinv  = 1.0f / (meanD + EPS_F);
  const float invSg = 16.0f / 3.0f;        // 1/sigma, sigma = BIN_MAX/K
  const float step  = 3.0f / 15.0f;        // linspace(0,3,16) spacing

  // ---- pass 2: recompute tiles, soft histogram in registers ----
  float h[KB];
#pragma unroll
  for (int k = 0; k < KB; ++k) h[k] = 0.0f;

  for (int idx = wave; idx < N_TILES; idx += 8) {
    int tm, tn; tile_coords(idx, tm, tn);
    v8f c = gram_tile(Hs, tm, tn, lo, hi);
    const float w   = (tm == tn) ? 1.0f : 2.0f;
    const float sqj = sq[tn * 16 + lo];
#pragma unroll
    for (int v = 0; v < 8; ++v) {
      float sqi = sq[tm * 16 + v + 8 * hi];
      float d2  = sqi + sqj - 2.0f * c[v];
      float d   = fast_sqrtf(fmaxf(d2, 0.0f) + 1e-12f);
      float dn  = d * rinv;
#pragma unroll
      for (int k = 0; k < KB; ++k) {
        float t = (dn - step * (float)k) * invSg;
        h[k] += w * __expf(-0.5f * t * t);
      }
    }
  }
#pragma unroll
  for (int k = 0; k < KB; ++k) atomicAdd(&red[1 + k], h[k]);
  __syncthreads();

  // ---- normalize signature, write to workspace ----
  if (tid < KB) {
    float tot = 0.0f;
#pragma unroll
    for (int k = 0; k < KB; ++k) tot += red[1 + k];
    sig[(size_t)b * KB + tid] = red[1 + tid] / (tot + EPS_F);
  }
}

__global__ void loss_kernel(const float* __restrict__ z1,
                            const float* __restrict__ z2,
                            const float* __restrict__ sig,
                            float* __restrict__ out) {
  extern __shared__ char smem_raw[];
  float* zn  = (float*)smem_raw;           // 256 * LP
  float* red = zn + 256 * LP;              // [0]=ntxent sum, [1]=mse sum

  const int t = threadIdx.x;               // 256 threads
  if (t < 2) red[t] = 0.0f;

  // normalize row t of z = [z1; z2]
  const float* zr = (t < NUM_GRAPHS) ? (z1 + (size_t)t * DF)
                                     : (z2 + (size_t)(t - NUM_GRAPHS) * DF);
  float s = 0.0f;
#pragma unroll 8
  for (int k = 0; k < DF; ++k) s += zr[k] * zr[k];
  const float inv = 1.0f / (fast_sqrtf(s) + EPS_F);
#pragma unroll 8
  for (int k = 0; k < DF; ++k) zn[t * LP + k] = zr[k] * inv;
  __syncthreads();

  // online softmax over row t of sim = (zn @ zn^T)/TEMP with masked diagonal
  const int label = (t + NUM_GRAPHS) & 255;
  float m = -3.0e38f, se = 0.0f, tgt = 0.0f;
  const float* zt = zn + t * LP;
  for (int j = 0; j < 2 * NUM_GRAPHS; ++j) {
    const float* zj = zn + j * LP;
    float dot = 0.0f;
#pragma unroll 16
    for (int k = 0; k < DF; ++k) dot += zt[k] * zj[k];
    float simv = (j == t) ? -1.0e9f : dot * 2.0f;   // / TEMP (0.5)
    if (j == label) tgt = simv;
    if (simv > m) { se = se * __expf(m - simv); m = simv; }
    se += __expf(simv - m);
  }
  atomicAdd(&red[0], -(tgt - m - __logf(se)));

  // topo MSE over signatures
  if (t < NUM_GRAPHS) {
    const float* s1 = sig + (size_t)t * KB;
    const float* s2 = sig + (size_t)(NUM_GRAPHS + t) * KB;
    float mse = 0.0f;
#pragma unroll
    for (int k = 0; k < KB; ++k) { float d = s1[k] - s2[k]; mse += d * d; }
    atomicAdd(&red[1], mse * (1.0f / (float)KB));
  }
  __syncthreads();

  if (t == 0) {
    float topo   = red[1] * (1.0f / (float)NUM_GRAPHS);
    float ntxent = red[0] * (1.0f / (float)(2 * NUM_GRAPHS));
    out[0] = LAMBDA_F * (topo + ntxent);
  }
}

extern "C" void kernel_launch(void* const* d_in, const int* in_sizes, int n_in,
                              void* d_out, int out_size, void* d_ws, size_t ws_size,
                              hipStream_t stream) {
  const float* H1 = (const float*)d_in[0];
  // d_in[1] = batch1 (unused: batch is arange // NPG by construction)
  const float* H2 = (const float*)d_in[2];
  // d_in[3] = batch2 (unused)
  const float* z1 = (const float*)d_in[4];
  const float* z2 = (const float*)d_in[5];
  float* out = (float*)d_out;
  float* sig = (float*)d_ws;               // 2*128*16 floats = 16 KB

  const size_t lds1 = (size_t)(NPG * LP + NPG + (KB + 1)) * sizeof(float);
  const size_t lds2 = (size_t)(256 * LP + 2) * sizeof(float);

  uts_sig_kernel<<<dim3(2 * NUM_GRAPHS), dim3(256), lds1, stream>>>(H1, H2, sig);
  loss_kernel<<<dim3(1), dim3(256), lds2, stream>>>(z1, z2, sig, out);
}